// PFBlock_57887569215569
// MI455X (gfx1250) — compile-verified
//
#include <hip/hip_runtime.h>
#include <hip/hip_bf16.h>

// ---------------- problem constants ----------------
#define BB   12
#define CC   64
#define KK   32
#define DD   32
#define NVOX 32768          // 32*32*32
#define MID  32
#define RED  16

typedef __attribute__((ext_vector_type(2))) float v2f;
typedef __attribute__((ext_vector_type(8))) float v8f;
typedef __attribute__((ext_vector_type(4))) unsigned int u32x4;
typedef __attribute__((ext_vector_type(4))) int i32x4;
typedef __attribute__((ext_vector_type(8))) int i32x8;

__device__ __forceinline__ v8f wmma_f32_4(v2f a, v2f b, v8f c) {
  // V_WMMA_F32_16X16X4_F32 : D = A(16x4 f32) * B(4x16 f32) + C(16x16 f32)
  return __builtin_amdgcn_wmma_f32_16x16x4_f32(false, a, false, b, (short)0, c,
                                               false, false);
}

#if __has_builtin(__builtin_amdgcn_tensor_load_to_lds) && \
    __has_builtin(__builtin_amdgcn_s_wait_tensorcnt)
#define USE_TDM 1
#else
#define USE_TDM 0
#endif

#if USE_TDM
// Issue one TDM descriptor: contiguous 1-D tile of `nfloats` f32,
// global -> LDS.  D# built per CDNA5 ISA §8 (groups 0/1; groups 2/3 and the
// extra group zeroed => tensor treated as <=2D).  Wave-level op, tracked by
// TENSORcnt.  6-arg builtin form (clang-23 / therock-10.0 toolchain).
__device__ __forceinline__ void tdm_load_f32(unsigned lds_addr,
                                             const float* gptr,
                                             unsigned nfloats) {
  const unsigned long long ga = (unsigned long long)(uintptr_t)gptr;
  u32x4 g0;
  g0[0] = 1u;                                   // count=1, user descriptor
  g0[1] = lds_addr;                             // LDS byte address
  g0[2] = (unsigned)(ga & 0xffffffffu);         // global_addr[31:0]
  g0[3] = (unsigned)((ga >> 32) & 0x01ffffffu)  // global_addr[56:32]
          | 0x80000000u;                        // type=2 ("image")
  i32x8 g1;
  g1[0] = (int)(2u << 16);                      // data_size=4B; no multicast
  g1[1] = (int)((nfloats & 0xffffu) << 16);     // tensor_dim0[15:0]
  g1[2] = (int)(((nfloats >> 16) & 0xffffu)     // tensor_dim0[31:16]
                | (1u << 16));                  // tensor_dim1 = 1
  g1[3] = (int)((nfloats & 0xffffu) << 16);     // tile_dim0 = nfloats (<=64Ki)
  g1[4] = 1;                                    // tile_dim1=1, tile_dim2=0
  g1[5] = (int)nfloats;                         // tensor_dim0_stride[31:0]
  g1[6] = 0;                                    // stride0 hi, stride1 lo
  g1[7] = 0;                                    // stride1 hi
  i32x4 gz4 = {0, 0, 0, 0};
  i32x8 gz8 = {0, 0, 0, 0, 0, 0, 0, 0};
  __builtin_amdgcn_tensor_load_to_lds(g0, g1, gz4, gz4, gz8, 0);
}
#endif

// ---------------- workspace layout (floats) ----------------
#define WS_RFPART 0                         // 16 * 24576  (split-K partials)
#define WS_OAPART (16 * 24576)              // 1536 * 96   (ortho partials)
#define WS_MSUM   (WS_OAPART + 1536 * 96)   // 32
#define WS_REG    (WS_MSUM + 32)            // 1
#define WS_WK     (WS_REG + 1)              // 384
#define WS_MOD    (WS_WK + 384)             // 24576
#define WS_RFN    (WS_MOD + 24576)          // 24576
#define WS_G2     (WS_RFN + 24576)          // 663552

// ---------------- output layout (floats) ----------------
#define OUT_RF    0
#define OUT_SPAT  24576
#define OUT_ZOUT  (24576 + BB * NVOX)
#define OUT_SC    (OUT_ZOUT + BB * CC * NVOX)

// ============================================================
// Kernel 1: rf partials via WMMA f32 16x16x4.
// out[c,k] = sum_x Z[b,c,x] * mask[k,x].  A = Z tile (M=c,K=x),
// B = masks tile (K=x, N=k).  Split-K over x (16 chunks of 2048).
// ============================================================
__global__ __launch_bounds__(32) void rf_wmma_kernel(
    const float* __restrict__ Z, const float* __restrict__ masks,
    float* __restrict__ rf_part) {
  const int lane = threadIdx.x, half = lane >> 4, lm = lane & 15;
  const int split = blockIdx.x;              // 0..15
  const int tile  = blockIdx.y;              // 0..7 : ct*2 + kt
  const int b     = blockIdx.z;              // 0..11
  const int ct = tile >> 1, kt = tile & 1;
  const int c0 = ct * 16, k0v = kt * 16;
  const int offA = half ? 2 : 0;

  const float* __restrict__ zrow = Z + (size_t)(b * CC + c0 + lm) * NVOX;
  const float* __restrict__ mrow = masks + (size_t)(k0v + lm) * NVOX;
  const int xbase = split * 2048;

  v8f acc = {0.f, 0.f, 0.f, 0.f, 0.f, 0.f, 0.f, 0.f};
  for (int s = 0; s < 512; ++s) {
    const int xb = xbase + 4 * s;
    v2f a, bv;
    a.x  = zrow[xb + offA];
    a.y  = zrow[xb + offA + 1];
    bv.x = mrow[xb + offA];
    bv.y = mrow[xb + offA + 1];
    acc = wmma_f32_4(a, bv, acc);
  }
  // D layout: VGPR i -> M = i + 8*half (=c), N = lm (=k)
  float* dst = rf_part + (size_t)split * 24576;
#pragma unroll
  for (int i = 0; i < 8; ++i) {
    const int c = c0 + i + 8 * half;
    const int k = k0v + lm;
    dst[(b * KK + k) * CC + c] = acc[i];
  }
}

// ============================================================
// Kernel 2a: msum[k] = sum_x masks[k,x]
// ============================================================
__global__ __launch_bounds__(256) void msum_kernel(
    const float* __restrict__ masks, float* __restrict__ msum) {
  __shared__ float red[256];
  const int k = blockIdx.x, t = threadIdx.x;
  float s = 0.f;
  for (int x = t; x < NVOX; x += 256) s += masks[(size_t)k * NVOX + x];
  red[t] = s;
  __syncthreads();
  for (int off = 128; off > 0; off >>= 1) {
    if (t < off) red[t] += red[t + off];
    __syncthreads();
  }
  if (t == 0) msum[k] = red[0];
}

// ============================================================
// Kernel 2b: deterministic split-K reduce + normalize; writes rf to
// both ws (for later stages) and d_out.
// ============================================================
__global__ __launch_bounds__(256) void rfnorm_kernel(
    const float* __restrict__ rf_part, const float* __restrict__ msum,
    float* __restrict__ rfn, float* __restrict__ rf_out) {
  const int idx = blockIdx.x * 256 + threadIdx.x;
  if (idx >= 24576) return;
  float s = 0.f;
  for (int sp = 0; sp < 16; ++sp) s += rf_part[(size_t)sp * 24576 + idx];
  const int k = (idx >> 6) & 31;
  const float v = s / (msum[k] + 1e-6f);
  rfn[idx] = v;
  rf_out[idx] = v;
}

// ============================================================
// Kernel 3: all the tiny stats in one block.
// MMD folded: D[k] = sum_{i,j} Wt[i,j] * mean_s kern[s,k,i,j], with
// Wt the class-pair coefficient matrix (independent of k and sigma).
// ============================================================
__device__ __forceinline__ float softplusf(float x) {
  return (x > 20.f) ? x : log1pf(__expf(x));
}

__global__ __launch_bounds__(256) void stats_kernel(
    const float* __restrict__ rfn, const int* __restrict__ labels,
    const float* __restrict__ ages, const float* __restrict__ prior,
    const float* __restrict__ m_w1, const float* __restrict__ m_b1,
    const float* __restrict__ m_w2, const float* __restrict__ m_b2,
    float* __restrict__ wk_out, float* __restrict__ mod_out,
    float* __restrict__ reg_out) {
  __shared__ int   sL[BB];
  __shared__ float scnt[3];
  __shared__ float sWt[BB * BB];
  __shared__ float sD[KK];
  __shared__ float sbase[KK];
  const int t = threadIdx.x;

  if (t < BB) sL[t] = labels[t];
  __syncthreads();
  if (t < 3) {
    int c = 0;
    for (int i = 0; i < BB; ++i) c += (sL[i] == t);
    scnt[t] = (float)c;
  }
  __syncthreads();
  if (t < BB * BB) {
    const int i = t / BB, j = t % BB;
    const int Li = sL[i], Lj = sL[j];
    const int pa[3] = {2, 1, 2}, pb[3] = {0, 0, 1};
    float w = 0.f;
    for (int p = 0; p < 3; ++p) {
      const float m = scnt[pa[p]], n = scnt[pb[p]];
      if (i != j && Li == pa[p] && Lj == pa[p]) w += 1.f / (m * (m - 1.f));
      if (i != j && Li == pb[p] && Lj == pb[p]) w += 1.f / (n * (n - 1.f));
      if (Li == pa[p] && Lj == pb[p]) w += -2.f / (m * n);
    }
    sWt[t] = w;
  }
  __syncthreads();
  if (t < KK) {
    const int k = t;
    float xx[BB];
#pragma unroll
    for (int i = 0; i < BB; ++i) {
      float s = 0.f;
      for (int c = 0; c < CC; ++c) {
        const float v = rfn[(i * KK + k) * CC + c];
        s += v * v;
      }
      xx[i] = s;
    }
    const float sig2inv[4] = {1.f / (2.f * 0.1f * 0.1f), 1.f / (2.f * 0.5f * 0.5f),
                              1.f / (2.f * 1.0f * 1.0f), 1.f / (2.f * 2.0f * 2.0f)};
    float D = 0.f;
    for (int i = 0; i < BB; ++i)
      for (int j = 0; j < BB; ++j) {
        const float w = sWt[i * BB + j];
        if (w != 0.f) {
          float ip = 0.f;
          for (int c = 0; c < CC; ++c)
            ip += rfn[(i * KK + k) * CC + c] * rfn[(j * KK + k) * CC + c];
          float d2 = xx[i] + xx[j] - 2.f * ip;
          d2 = d2 > 0.f ? d2 : 0.f;
          float kk = 0.f;
#pragma unroll
          for (int s4 = 0; s4 < 4; ++s4) kk += __expf(-d2 * sig2inv[s4]);
          D += w * (0.25f * kk);
        }
      }
    sD[k] = D;
  }
  __syncthreads();
  if (t == 0) {
    float mx = sD[0];
    for (int k = 1; k < KK; ++k) mx = fmaxf(mx, sD[k]);
    mx = fmaxf(mx, 1e-6f);
    float nm[KK];
    for (int k = 0; k < KK; ++k) {
      const float v = sD[k] / mx;
      nm[k] = v > 0.f ? v : 0.f;
    }
    float mp = prior[0], mq = nm[0];
    for (int k = 1; k < KK; ++k) { mp = fmaxf(mp, prior[k]); mq = fmaxf(mq, nm[k]); }
    float sp = 0.f, sq = 0.f;
    for (int k = 0; k < KK; ++k) { sp += __expf(prior[k] - mp); sq += __expf(nm[k] - mq); }
    const float lsep = mp + __logf(sp), lseq = mq + __logf(sq);
    float reg = 0.f;
    for (int k = 0; k < KK; ++k) {
      const float lp = prior[k] - lsep, lq = nm[k] - lseq;
      reg += 0.5f * __expf(lq) * (lq - lp) + 0.5f * __expf(lp) * (lp - lq);
    }
    *reg_out = reg * (1.f / (float)KK);
    for (int k = 0; k < KK; ++k) sbase[k] = 0.7f * prior[k] + 0.3f * nm[k];
  }
  __syncthreads();
  if (t < BB) {
    const float af = 1.f + softplusf(0.1f * (ages[t] - 50.f));
    float row[KK];
    float mx = -1e30f;
#pragma unroll
    for (int k = 0; k < KK; ++k) {
      float v = sbase[k] * af;
      v = fminf(fmaxf(v, 0.f), 2.f);
      row[k] = v;
      mx = fmaxf(mx, v);
    }
    float s = 0.f;
#pragma unroll
    for (int k = 0; k < KK; ++k) { row[k] = __expf(row[k] - mx); s += row[k]; }
    const float inv = 1.f / s;
#pragma unroll
    for (int k = 0; k < KK; ++k) wk_out[t * KK + k] = row[k] * inv;
  }
  __syncthreads();
  for (int idx = t; idx < BB * KK; idx += 256) {
    const float* r = rfn + idx * CC;
    float h[RED];
#pragma unroll
    for (int rr = 0; rr < RED; ++rr) {
      float s = m_b1[rr];
      for (int c = 0; c < CC; ++c) s += r[c] * m_w1[rr * CC + c];
      h[rr] = s > 0.f ? s : 0.f;
    }
    for (int c = 0; c < CC; ++c) {
      float s = m_b2[c];
#pragma unroll
      for (int rr = 0; rr < RED; ++rr) s += h[rr] * m_w2[c * RED + rr];
      mod_out[idx * CC + c] = softplusf(s);
    }
  }
}

// ============================================================
// Kernel 4a: G2[b,o,t,k] = sum_c region_w[o,c,t] * rf[b,k,c]
// ============================================================
__global__ __launch_bounds__(256) void g2_kernel(
    const float* __restrict__ region_w, const float* __restrict__ rfn,
    float* __restrict__ G2) {
  const int idx = blockIdx.x * 256 + threadIdx.x;   // 663552 total, exact
  const int k = idx & 31;
  const int tmp = idx >> 5;
  const int t = tmp % 27;
  const int tmp2 = tmp / 27;
  const int o = tmp2 & 63;
  const int b = tmp2 >> 6;
  float s = 0.f;
  const float* rr = rfn + (b * KK + k) * CC;
  for (int c = 0; c < CC; ++c) s += region_w[(o * CC + c) * 27 + t] * rr[c];
  G2[idx] = s;
}

// ============================================================
// Kernel 4b: conv as implicit GEMM over (t,k) via WMMA f32 16x16x4.
// Zrec[b,o,x] = sum_{t,k} G2[b,o,t,k] * mask[k, shift_t(x)]
// Writes Zrec directly into the Z_out region of d_out.
// ============================================================
__global__ __launch_bounds__(32) void conv_wmma_kernel(
    const float* __restrict__ G2, const float* __restrict__ masks,
    float* __restrict__ zout) {
  const int lane = threadIdx.x, half = lane >> 4, lm = lane & 15;
  const int x0 = blockIdx.x * 16;           // 2048 tiles, within one W-row
  const int o0 = blockIdx.y * 16;           // 4 tiles
  const int b  = blockIdx.z;                // 12
  const int d0 = x0 >> 10, h0 = (x0 >> 5) & 31, w0 = x0 & 31;
  const int w  = w0 + lm;
  const int offA = half ? 2 : 0;

  const float* __restrict__ g2row = G2 + (size_t)((b * CC + o0 + lm) * 27) * KK;

  v8f acc = {0.f, 0.f, 0.f, 0.f, 0.f, 0.f, 0.f, 0.f};
  for (int t = 0; t < 27; ++t) {
    const int td = t / 9 - 1, th = (t / 3) % 3 - 1, tw = t % 3 - 1;
    const int dn = d0 + td, hn = h0 + th, wn = w + tw;
    const bool valid = ((unsigned)dn < 32u) & ((unsigned)hn < 32u) & ((unsigned)wn < 32u);
    const int xn = valid ? ((dn << 10) + (hn << 5) + wn) : 0;
    const float vf = valid ? 1.f : 0.f;
    const float* __restrict__ g2t = g2row + t * KK + offA;
    if (t < 26) __builtin_prefetch(g2row + (t + 1) * KK, 0, 0);
#pragma unroll
    for (int k0 = 0; k0 < KK; k0 += 4) {
      v2f a, bv;
      a.x = g2t[k0];
      a.y = g2t[k0 + 1];
      const int ka = k0 + offA;
      bv.x = masks[(size_t)ka * NVOX + xn] * vf;
      bv.y = masks[(size_t)(ka + 1) * NVOX + xn] * vf;
      acc = wmma_f32_4(a, bv, acc);
    }
  }
  const int x = x0 + lm;
#pragma unroll
  for (int i = 0; i < 8; ++i) {
    const int o = o0 + i + 8 * half;
    zout[(size_t)(b * CC + o) * NVOX + x] = acc[i];
  }
}

// ============================================================
// Kernel 5: fused per-voxel epilogue.
//  s = struct_w@Z, n = noise_w@Z, Z_clean = up_w@s,
//  spatial = Wk . mask, mm = mod . mask,
//  Z_out = Z_clean + 0.1 * Zrec * mm * spatial  (Zrec read in place)
//  + deterministic wave-level ortho partials.
// Weight staging into LDS via the Tensor Data Mover (4 D# descriptors,
// issued by wave 0 only -- TDM ignores EXEC, so the guard must be
// wave-uniform), completed with s_wait_tensorcnt + workgroup barrier.
// ============================================================
__global__ __launch_bounds__(256) void fuse_kernel(
    const float* __restrict__ Z, const float* __restrict__ masks,
    const float* __restrict__ struct_w, const float* __restrict__ noise_w,
    const float* __restrict__ up_w, const float* __restrict__ wk,
    const float* __restrict__ mod, float* __restrict__ spatial_out,
    float* __restrict__ zout, float* __restrict__ oaPart) {
  __shared__ float ssw[MID * CC];
  __shared__ float snw[MID * CC];
  __shared__ float sup[CC * MID];
  __shared__ float smod[KK * CC];
  __shared__ float swk[KK];
  __shared__ float swred[8 * 96];

  const int t = threadIdx.x;
  const int b = blockIdx.y;
#if USE_TDM
  if (t < 32) {  // wave 0 only (uniform branch: TDM issues regardless of EXEC)
    tdm_load_f32((unsigned)(uintptr_t)&ssw[0],  struct_w,      MID * CC);
    tdm_load_f32((unsigned)(uintptr_t)&snw[0],  noise_w,       MID * CC);
    tdm_load_f32((unsigned)(uintptr_t)&sup[0],  up_w,          CC * MID);
    tdm_load_f32((unsigned)(uintptr_t)&smod[0], mod + b * KK * CC, KK * CC);
    __builtin_amdgcn_s_wait_tensorcnt(0);
  }
#else
  for (int i = t; i < MID * CC; i += 256) {
    ssw[i] = struct_w[i];
    snw[i] = noise_w[i];
    sup[i] = up_w[i];
    smod[i] = mod[b * KK * CC + i];
  }
#endif
  if (t < KK) swk[t] = wk[b * KK + t];
  __syncthreads();

  const int x = blockIdx.x * 256 + t;
  float s[MID], nn[MID];
#pragma unroll
  for (int m = 0; m < MID; ++m) { s[m] = 0.f; nn[m] = 0.f; }

  const float* __restrict__ zb = Z + (size_t)b * CC * NVOX + x;
  for (int c = 0; c < CC; ++c) {
    const float zc = zb[(size_t)c * NVOX];
#pragma unroll
    for (int m = 0; m < MID; ++m) {
      s[m]  = fmaf(ssw[m * CC + c], zc, s[m]);
      nn[m] = fmaf(snw[m * CC + c], zc, nn[m]);
    }
  }

  // ortho partial sums: wave32 shuffle tree -> per-wave, then fixed-order.
  const int wid = t >> 5, ln = t & 31;
#pragma unroll
  for (int m = 0; m < MID; ++m) {
    float a0 = s[m] * nn[m], a1 = s[m] * s[m], a2 = nn[m] * nn[m];
    for (int off = 16; off > 0; off >>= 1) {
      a0 += __shfl_down(a0, off);
      a1 += __shfl_down(a1, off);
      a2 += __shfl_down(a2, off);
    }
    if (ln == 0) {
      swred[wid * 96 + m * 3 + 0] = a0;
      swred[wid * 96 + m * 3 + 1] = a1;
      swred[wid * 96 + m * 3 + 2] = a2;
    }
  }

  float mv[KK];
#pragma unroll
  for (int k = 0; k < KK; ++k) mv[k] = masks[(size_t)k * NVOX + x];
  float spat = 0.f;
#pragma unroll
  for (int k = 0; k < KK; ++k) spat = fmaf(swk[k], mv[k], spat);
  spatial_out[b * NVOX + x] = spat;

  float* __restrict__ zob = zout + (size_t)b * CC * NVOX + x;
  for (int o = 0; o < CC; ++o) {
    float zc = 0.f, mm = 0.f;
#pragma unroll
    for (int m = 0; m < MID; ++m) zc = fmaf(sup[o * MID + m], s[m], zc);
#pragma unroll
    for (int k = 0; k < KK; ++k) mm = fmaf(smod[k * CC + o], mv[k], mm);
    const float zrec = zob[(size_t)o * NVOX];
    zob[(size_t)o * NVOX] = zc + 0.1f * (zrec * mm * spat);
  }

  __syncthreads();
  if (t < 96) {
    float a = 0.f;
    for (int wv = 0; wv < 8; ++wv) a += swred[wv * 96 + t];
    oaPart[(size_t)(b * 128 + blockIdx.x) * 96 + t] = a;
  }
}

// ============================================================
// Kernel 6: ortho = mean |cos|; scalar out = reg + ortho.
// ============================================================
__global__ __launch_bounds__(512) void scalar_kernel(
    const float* __restrict__ oaPart, const float* __restrict__ reg,
    float* __restrict__ out) {
  __shared__ float red[512];
  const int t = threadIdx.x;
  float v = 0.f;
  if (t < BB * MID) {
    const int b = t >> 5, m = t & 31;
    float d = 0.f, a = 0.f, c2 = 0.f;
    for (int ch = 0; ch < 128; ++ch) {
      const float* p = oaPart + (size_t)(b * 128 + ch) * 96 + m * 3;
      d += p[0];
      a += p[1];
      c2 += p[2];
    }
    const float na = fmaxf(sqrtf(a), 1e-8f), nb = fmaxf(sqrtf(c2), 1e-8f);
    v = fabsf(d / (na * nb));
  }
  red[t] = v;
  __syncthreads();
  for (int off = 256; off > 0; off >>= 1) {
    if (t < off) red[t] += red[t + off];
    __syncthreads();
  }
  if (t == 0) out[0] = red[0] * (1.f / 384.f) + reg[0];
}

// ============================================================
extern "C" void kernel_launch(void* const* d_in, const int* in_sizes, int n_in,
                              void* d_out, int out_size, void* d_ws,
                              size_t ws_size, hipStream_t stream) {
  (void)in_sizes; (void)n_in; (void)out_size; (void)ws_size;
  const float* Z        = (const float*)d_in[0];
  const float* masks    = (const float*)d_in[1];
  const int*   labels   = (const int*)d_in[2];
  const float* ages     = (const float*)d_in[3];
  const float* prior    = (const float*)d_in[4];
  const float* struct_w = (const float*)d_in[5];
  const float* noise_w  = (const float*)d_in[6];
  const float* up_w     = (const float*)d_in[7];
  const float* region_w = (const float*)d_in[8];
  const float* m_w1     = (const float*)d_in[9];
  const float* m_b1     = (const float*)d_in[10];
  const float* m_w2     = (const float*)d_in[11];
  const float* m_b2     = (const float*)d_in[12];

  float* out = (float*)d_out;
  float* ws  = (float*)d_ws;
  float* rf_part = ws + WS_RFPART;
  float* oaPart  = ws + WS_OAPART;
  float* msum    = ws + WS_MSUM;
  float* regbuf  = ws + WS_REG;
  float* wk      = ws + WS_WK;
  float* mod     = ws + WS_MOD;
  float* rfn     = ws + WS_RFN;
  float* G2      = ws + WS_G2;

  rf_wmma_kernel<<<dim3(16, 8, BB), 32, 0, stream>>>(Z, masks, rf_part);
  msum_kernel<<<KK, 256, 0, stream>>>(masks, msum);
  rfnorm_kernel<<<96, 256, 0, stream>>>(rf_part, msum, rfn, out + OUT_RF);
  stats_kernel<<<1, 256, 0, stream>>>(rfn, labels, ages, prior, m_w1, m_b1,
                                      m_w2, m_b2, wk, mod, regbuf);
  g2_kernel<<<2592, 256, 0, stream>>>(region_w, rfn, G2);
  conv_wmma_kernel<<<dim3(2048, 4, BB), 32, 0, stream>>>(G2, masks,
                                                         out + OUT_ZOUT);
  fuse_kernel<<<dim3(128, BB), 256, 0, stream>>>(Z, masks, struct_w, noise_w,
                                                 up_w, wk, mod, out + OUT_SPAT,
                                                 out + OUT_ZOUT, oaPart);
  scalar_kernel<<<1, 512, 0, stream>>>(oaPart, regbuf, out + OUT_SC);
}